// IPATransformer_1915555414556
// MI455X (gfx1250) — compile-verified
//
#include <hip/hip_runtime.h>

// ---------------------------------------------------------------------------
// IPA transformer forward for MI455X (gfx1250, wave32).
// All matrix contractions use V_WMMA_F32_16X16X4_F32 (fp32 in/out) because the
// workload is HBM-bandwidth-bound (pair tensor traffic dominates), so fp32
// WMMA is free and preserves reference numerics.
// ---------------------------------------------------------------------------

#define BB    2
#define NN    512
#define DIMC  512
#define HH    12
#define SKC   16
#define SVC   16
#define PKC   4
#define PVC   8
#define PRC   128
#define DEPTH 4
#define OIN   2112   // H*(SV + PV*3 + PV + PR) = 12*176

typedef __attribute__((ext_vector_type(2))) float v2f;
typedef __attribute__((ext_vector_type(8))) float v8f;

static constexpr float SCALE_S_C    = 0.14433756729740643f;  // (3*16)^-0.5
static constexpr float SCALE_P_C    = 0.13608276348795434f;  // (3*4*4.5)^-0.5
static constexpr float SCALE_PAIR_C = 0.57735026918962576f;  // 3^-0.5
static constexpr float NEG_MAX      = -3.402823466e+38f;

// ---------------------------------------------------------------------------
// Generic batched GEMM: C = A @ B (+bias) (optional ReLU), fp32 WMMA 16x16x4.
// One wave computes one 16x16 tile of C. Batch index z = blockIdx.y; per-z
// pointer offset = (z/inner)*stride_outer + (z%inner)*stride_inner for each
// of A, B, C.
//
// Out-of-range handling WITHOUT per-lane load predication: clamp the A row
// index to M-1 and the B column index to Nc-1 and load unconditionally.
// Garbage in A row m only affects D row m; garbage in B col n only affects
// D col n — and those D entries are never stored. This keeps the inner loop
// free of exec-mask branches (the previous version serialized on them).
//
// VGPR layouts per CDNA5 ISA 7.12.2:
//   A(m,k): lane = m + 16*(k/2), component = k%2   -> one v2f (b64) load
//   B(k,n): lane = n + 16*(k/2), component = k%2
//   D(m,n): vgpr  = m%8, lane  = n + 16*(m/8)
// K must be a multiple of 4 (call sites: 32/128/512/2112). All A strides and
// base offsets are even, so the v2f loads are 8-byte aligned.
// ---------------------------------------------------------------------------
__global__ void __launch_bounds__(128)
k_gemm(const float* __restrict__ A, const float* __restrict__ Bm,
       float* __restrict__ C, const float* __restrict__ bias,
       int M, int Nc, int K, int lda, int ldb, int ldc,
       long sAo, long sAi, int ain,
       long sBo, long sBi, int bin,
       long sCo, long sCi, int cin,
       int transB, int relu)
{
    int z = blockIdx.y;
    const float* Ab = A  + (long)(z / ain) * sAo + (long)(z % ain) * sAi;
    const float* Bb = Bm + (long)(z / bin) * sBo + (long)(z % bin) * sBi;
    float*       Cb = C  + (long)(z / cin) * sCo + (long)(z % cin) * sCi;

    int tilesN = (Nc + 15) >> 4;
    int tilesM = (M  + 15) >> 4;
    int tile = blockIdx.x * 4 + (int)(threadIdx.x >> 5);
    if (tile >= tilesM * tilesN) return;
    int tm = tile / tilesN;
    int tn = tile - tm * tilesN;
    int m0 = tm << 4, n0 = tn << 4;

    int lane = (int)(threadIdx.x & 31);
    int lgrp = lane >> 4;     // 0: K 0..1 half, 1: K 2..3 half
    int lidx = lane & 15;     // m for A, n for B/D

    int am  = m0 + lidx; if (am > M - 1)  am = M - 1;   // clamp (see header)
    int bn0 = n0 + lidx;
    int bn  = bn0 > Nc - 1 ? Nc - 1 : bn0;

    const float* Arow = Ab + (long)am * lda;
    const float* Brow = Bb + (long)bn * ldb;
    int koff = lgrp << 1;

    v8f acc = {0.f, 0.f, 0.f, 0.f, 0.f, 0.f, 0.f, 0.f};

    if (transB) {
        // both operands are consecutive float pairs -> b64 loads
        v2f a = *(const v2f*)(Arow + koff);
        v2f b = *(const v2f*)(Brow + koff);
        for (int k0 = 4; k0 < K; k0 += 4) {
            v2f an = *(const v2f*)(Arow + k0 + koff);
            v2f bnx = *(const v2f*)(Brow + k0 + koff);
            acc = __builtin_amdgcn_wmma_f32_16x16x4_f32(false, a, false, b,
                                                        (short)0, acc, false, false);
            a = an; b = bnx;
        }
        acc = __builtin_amdgcn_wmma_f32_16x16x4_f32(false, a, false, b,
                                                    (short)0, acc, false, false);
    } else {
        const float* Bcol = Bb + bn;
        v2f a = *(const v2f*)(Arow + koff);
        v2f b;
        b.x = Bcol[(long)koff * ldb];
        b.y = Bcol[(long)(koff + 1) * ldb];
        for (int k0 = 4; k0 < K; k0 += 4) {
            v2f an = *(const v2f*)(Arow + k0 + koff);
            v2f bnx;
            bnx.x = Bcol[(long)(k0 + koff) * ldb];
            bnx.y = Bcol[(long)(k0 + koff + 1) * ldb];
            acc = __builtin_amdgcn_wmma_f32_16x16x4_f32(false, a, false, b,
                                                        (short)0, acc, false, false);
            a = an; b = bnx;
        }
        acc = __builtin_amdgcn_wmma_f32_16x16x4_f32(false, a, false, b,
                                                    (short)0, acc, false, false);
    }

    if (bn0 < Nc) {
        float bv = bias ? bias[bn0] : 0.f;
#pragma unroll
        for (int r = 0; r < 8; ++r) {
            int cm = m0 + (lgrp << 3) + r;
            if (cm < M) {
                float v = acc[r] + bv;
                if (relu) v = fmaxf(v, 0.f);
                Cb[(long)cm * ldc + bn0] = v;
            }
        }
    }
}

// -------------------- init: x copy, identity pose --------------------------
__global__ void k_init(const float* __restrict__ xin, float* __restrict__ x,
                       float* __restrict__ quats, float* __restrict__ trans, int total)
{
    int i = blockIdx.x * blockDim.x + threadIdx.x;
    if (i < total) x[i] = xin[i];
    if (i < BB * NN) {
        quats[i * 4 + 0] = 1.f; quats[i * 4 + 1] = 0.f;
        quats[i * 4 + 2] = 0.f; quats[i * 4 + 3] = 0.f;
        trans[i * 3 + 0] = 0.f; trans[i * 3 + 1] = 0.f; trans[i * 3 + 2] = 0.f;
    }
}

// -------------------- quats -> rotation matrices ---------------------------
__global__ void k_compute_R(const float* __restrict__ quats, float* __restrict__ R)
{
    int t = blockIdx.x * blockDim.x + threadIdx.x;
    if (t >= BB * NN) return;
    float r = quats[t * 4], i = quats[t * 4 + 1], j = quats[t * 4 + 2], k = quats[t * 4 + 3];
    float s = 2.f / (r * r + i * i + j * j + k * k);
    float* Rr = R + (long)t * 9;
    Rr[0] = 1.f - s * (j * j + k * k); Rr[1] = s * (i * j - k * r); Rr[2] = s * (i * k + j * r);
    Rr[3] = s * (i * j + k * r); Rr[4] = 1.f - s * (i * i + k * k); Rr[5] = s * (j * k - i * r);
    Rr[6] = s * (i * k - j * r); Rr[7] = s * (j * k + i * r); Rr[8] = 1.f - s * (i * i + j * j);
}

// -------------------- augmented q/k rows for the logits GEMM ---------------
// aq = [qs*SCALE_S, c*qp_glob(12), 1, 0,0,0]; ak = [ks, kp_glob(12), -0.5c*k2, 0,0,0]
// (the -0.5c*q2 term is a per-row constant -> softmax-invariant, dropped)
__global__ void k_build_aqak(const float* __restrict__ qs, const float* __restrict__ ks,
                             const float* __restrict__ qp, const float* __restrict__ kp,
                             const float* __restrict__ R, const float* __restrict__ tr,
                             const float* __restrict__ ptw,
                             float* __restrict__ AQ, float* __restrict__ AK)
{
    int t = blockIdx.x * blockDim.x + threadIdx.x;
    if (t >= BB * HH * NN) return;
    int n = t % NN;
    int h = (t / NN) % HH;
    int b = t / (NN * HH);
    long rn = (long)(b * NN + n);
    const float* qsr = qs + rn * (HH * SKC) + h * SKC;
    const float* ksr = ks + rn * (HH * SKC) + h * SKC;
    const float* qpr = qp + rn * (HH * PKC * 3) + h * (PKC * 3);
    const float* kpr = kp + rn * (HH * PKC * 3) + h * (PKC * 3);
    const float* Rr = R + rn * 9;
    const float* tv = tr + rn * 3;
    float ch = SCALE_P_C * log1pf(expf(ptw[h]));   // SCALE_P * softplus(pt_w)
    float* aq = AQ + (long)t * 32;
    float* ak = AK + (long)t * 32;
#pragma unroll
    for (int s = 0; s < SKC; ++s) { aq[s] = qsr[s] * SCALE_S_C; ak[s] = ksr[s]; }
    float k2 = 0.f;
#pragma unroll
    for (int d = 0; d < PKC; ++d) {
        float q0 = qpr[d * 3], q1 = qpr[d * 3 + 1], q2v = qpr[d * 3 + 2];
        float kx = kpr[d * 3], ky = kpr[d * 3 + 1], kz = kpr[d * 3 + 2];
#pragma unroll
        for (int r = 0; r < 3; ++r) {   // out_r = sum_c v_c * R[c][r] + t_r
            float gq = q0 * Rr[r] + q1 * Rr[3 + r] + q2v * Rr[6 + r] + tv[r];
            float gk = kx * Rr[r] + ky * Rr[3 + r] + kz * Rr[6 + r] + tv[r];
            aq[16 + d * 3 + r] = ch * gq;
            ak[16 + d * 3 + r] = gk;
            k2 += gk * gk;
        }
    }
    aq[28] = 1.f; aq[29] = 0.f; aq[30] = 0.f; aq[31] = 0.f;
    ak[28] = -0.5f * ch * k2; ak[29] = 0.f; ak[30] = 0.f; ak[31] = 0.f;
}

// -------------------- value points to global frame -------------------------
__global__ void k_transform_vp(const float* __restrict__ vp, const float* __restrict__ R,
                               const float* __restrict__ tr, float* __restrict__ vg)
{
    int t = blockIdx.x * blockDim.x + threadIdx.x;
    if (t >= BB * NN * HH * PVC) return;
    int d = t % PVC;
    int h = (t / PVC) % HH;
    long rn = t / (PVC * HH);      // b*NN + n
    const float* in = vp + rn * (HH * PVC * 3) + (h * PVC + d) * 3;
    float* out = vg + rn * (HH * PVC * 3) + (h * PVC + d) * 3;
    const float* Rr = R + rn * 9;
    const float* tv = tr + rn * 3;
    float v0 = in[0], v1 = in[1], v2 = in[2];
#pragma unroll
    for (int r = 0; r < 3; ++r)
        out[r] = v0 * Rr[r] + v1 * Rr[3 + r] + v2 * Rr[6 + r] + tv[r];
}

// -------------------- softmax row (adds pair bias + mask) ------------------
__global__ void __launch_bounds__(256)
k_softmax(float* __restrict__ logits, const float* __restrict__ pbias,
          const int* __restrict__ mask)
{
    int row = blockIdx.x;                 // (b*HH + h)*NN + i
    int i = row & (NN - 1);
    int h = (row >> 9) % HH;
    int b = row / (NN * HH);
    float* lr = logits + (long)row * NN;
    const float* pb = pbias + ((long)(b * NN + i)) * NN * HH + h;
    bool mi = mask[b * NN + i] != 0;
    int t = threadIdx.x;
    int j0 = t, j1 = t + 256;

    float v0 = lr[j0] + pb[(long)j0 * HH] * SCALE_PAIR_C;
    if (!(mi && mask[b * NN + j0] != 0)) v0 = NEG_MAX;
    float v1 = lr[j1] + pb[(long)j1 * HH] * SCALE_PAIR_C;
    if (!(mi && mask[b * NN + j1] != 0)) v1 = NEG_MAX;

    __shared__ float red[256];
    red[t] = fmaxf(v0, v1);
    __syncthreads();
    for (int s = 128; s > 0; s >>= 1) {
        if (t < s) red[t] = fmaxf(red[t], red[t + s]);
        __syncthreads();
    }
    float mx = red[0];
    __syncthreads();
    float e0 = expf(v0 - mx);
    float e1 = expf(v1 - mx);
    red[t] = e0 + e1;
    __syncthreads();
    for (int s = 128; s > 0; s >>= 1) {
        if (t < s) red[t] += red[t + s];
        __syncthreads();
    }
    float inv = 1.f / red[0];
    lr[j0] = e0 * inv;
    lr[j1] = e1 * inv;
}

// ------------- rpt back to local frame + norms, into concat buffer ---------
__global__ void k_finalize_rpt(const float* __restrict__ rg, const float* __restrict__ R,
                               const float* __restrict__ tr, float* __restrict__ o)
{
    int t = blockIdx.x * blockDim.x + threadIdx.x;
    if (t >= BB * NN * HH * PVC) return;
    int d = t % PVC;
    int h = (t / PVC) % HH;
    long rn = t / (PVC * HH);          // b*NN + n
    int n = (int)(rn % NN);
    int b = (int)(rn / NN);
    const float* in = rg + ((long)(b * HH + h) * NN + n) * (PVC * 3) + d * 3;
    const float* Rr = R + rn * 9;
    const float* tv = tr + rn * 3;
    float v0 = in[0] - tv[0], v1 = in[1] - tv[1], v2 = in[2] - tv[2];
    float l0 = Rr[0] * v0 + Rr[1] * v1 + Rr[2] * v2;   // out_r = sum_c R[r][c] v_c
    float l1 = Rr[3] * v0 + Rr[4] * v1 + Rr[5] * v2;
    float l2 = Rr[6] * v0 + Rr[7] * v1 + Rr[8] * v2;
    float* orow = o + rn * OIN;
    orow[HH * SVC + (h * PVC + d) * 3 + 0] = l0;
    orow[HH * SVC + (h * PVC + d) * 3 + 1] = l1;
    orow[HH * SVC + (h * PVC + d) * 3 + 2] = l2;
    orow[HH * SVC + HH * PVC * 3 + h * PVC + d] = sqrtf(l0 * l0 + l1 * l1 + l2 * l2 + 1e-8f);
}

// -------------------- x = LN(a + x) * g + b --------------------------------
__global__ void __launch_bounds__(256)
k_layernorm(const float* __restrict__ a, float* __restrict__ x,
            const float* __restrict__ g, const float* __restrict__ bta)
{
    __shared__ float y[DIMC];
    __shared__ float red[256];
    int row = blockIdx.x;
    int t = threadIdx.x;
    const float* ar = a + (long)row * DIMC;
    float* xr = x + (long)row * DIMC;
    float s = 0.f;
    for (int c = t; c < DIMC; c += 256) { float v = ar[c] + xr[c]; y[c] = v; s += v; }
    red[t] = s; __syncthreads();
    for (int st = 128; st > 0; st >>= 1) { if (t < st) red[t] += red[t + st]; __syncthreads(); }
    float m = red[0] * (1.f / DIMC);
    __syncthreads();
    s = 0.f;
    for (int c = t; c < DIMC; c += 256) { float d = y[c] - m; s += d * d; }
    red[t] = s; __syncthreads();
    for (int st = 128; st > 0; st >>= 1) { if (t < st) red[t] += red[t + st]; __syncthreads(); }
    float inv = rsqrtf(red[0] * (1.f / DIMC) + 1e-5f);
    for (int c = t; c < DIMC; c += 256) xr[c] = (y[c] - m) * inv * g[c] + bta[c];
}

// -------------------- quaternion + translation update ----------------------
__global__ void k_update_pose(const float* __restrict__ upd, const float* __restrict__ R,
                              float* __restrict__ quats, float* __restrict__ trans)
{
    int t = blockIdx.x * blockDim.x + threadIdx.x;
    if (t >= BB * NN) return;
    const float* u = upd + (long)t * 6;
    float* q = quats + (long)t * 4;
    float* tv = trans + (long)t * 3;
    const float* Rr = R + (long)t * 9;
    float aw = q[0], ax = q[1], ay = q[2], az = q[3];
    float bx = u[0], by = u[1], bz = u[2];          // qu = (1, bx, by, bz)
    float w  = aw - ax * bx - ay * by - az * bz;
    float xo = aw * bx + ax + ay * bz - az * by;
    float yo = aw * by - ax * bz + ay + az * bx;
    float zo = aw * bz + ax * by - ay * bx + az;
    if (w < 0.f) { w = -w; xo = -xo; yo = -yo; zo = -zo; }
    q[0] = w; q[1] = xo; q[2] = yo; q[3] = zo;
    float u3 = u[3], u4 = u[4], u5 = u[5];
#pragma unroll
    for (int r = 0; r < 3; ++r)                     // t_r += sum_c u_c * R[c][r]
        tv[r] += u3 * Rr[r] + u4 * Rr[3 + r] + u5 * Rr[6 + r];
}

// -------------------- final concat [x, trans, quats] -----------------------
__global__ void k_write_out(const float* __restrict__ x, const float* __restrict__ trans,
                            const float* __restrict__ quats, float* __restrict__ out)
{
    int t = blockIdx.x * blockDim.x + threadIdx.x;
    const int W = DIMC + 7;
    if (t >= BB * NN * W) return;
    int col = t % W;
    long rn = t / W;
    float v;
    if (col < DIMC)          v = x[rn * DIMC + col];
    else if (col < DIMC + 3) v = trans[rn * 3 + (col - DIMC)];
    else                     v = quats[rn * 4 + (col - DIMC - 3)];
    out[t] = v;
}

// ---------------------------------------------------------------------------
static void launch_gemm(hipStream_t st,
    const float* A, const float* Bm, float* C, const float* bias,
    int M, int Nc, int K, int lda, int ldb, int ldc,
    long sAo, long sAi, int ain,
    long sBo, long sBi, int bin,
    long sCo, long sCi, int cin,
    int batch, int transB, int relu)
{
    int tiles = ((M + 15) / 16) * ((Nc + 15) / 16);
    dim3 grid((unsigned)((tiles + 3) / 4), (unsigned)batch, 1);
    k_gemm<<<grid, dim3(128, 1, 1), 0, st>>>(A, Bm, C, bias, M, Nc, K, lda, ldb, ldc,
        sAo, sAi, ain, sBo, sBi, bin, sCo, sCi, cin, transB, relu);
}

extern "C" void kernel_launch(void* const* d_in, const int* in_sizes, int n_in,
                              void* d_out, int out_size, void* d_ws, size_t ws_size,
                              hipStream_t stream)
{
    (void)in_sizes; (void)n_in; (void)out_size; (void)ws_size;

    const float* xin   = (const float*)d_in[0];
    const float* pair  = (const float*)d_in[1];
    const int*   mask  = (const int*)d_in[2];
    const float* Wsq   = (const float*)d_in[3];
    const float* Wsk   = (const float*)d_in[4];
    const float* Wsv   = (const float*)d_in[5];
    const float* Wpq   = (const float*)d_in[6];
    const float* Wpk   = (const float*)d_in[7];
    const float* Wpv   = (const float*)d_in[8];
    const float* ptw   = (const float*)d_in[9];
    const float* Wpair = (const float*)d_in[10];
    const float* bpair = (const float*)d_in[11];
    const float* Wo    = (const float*)d_in[12];
    const float* bo    = (const float*)d_in[13];
    const float* ln1g  = (const float*)d_in[14];
    const float* ln1b  = (const float*)d_in[15];
    const float* ln2g  = (const float*)d_in[16];
    const float* ln2b  = (const float*)d_in[17];
    const float* W1    = (const float*)d_in[18];
    const float* b1    = (const float*)d_in[19];
    const float* W2    = (const float*)d_in[20];
    const float* b2    = (const float*)d_in[21];
    const float* W3    = (const float*)d_in[22];
    const float* b3    = (const float*)d_in[23];
    const float* Wupd  = (const float*)d_in[24];
    const float* bupd  = (const float*)d_in[25];

    float* p = (float*)d_ws;
    size_t off = 0;
    auto alloc = [&](size_t nf) { float* r = p + off; off += nf; return r; };

    const int M = BB * NN;                       // 1024 rows of x
    float* x     = alloc((size_t)M * DIMC);
    float* quats = alloc((size_t)M * 4);
    float* trans = alloc((size_t)M * 3 + 1);     // keep following buffers even-aligned
    float* Rbuf  = alloc((size_t)M * 9 + 1);
    float* qs_b  = alloc((size_t)M * HH * SKC);
    float* ks_b  = alloc((size_t)M * HH * SKC);
    float* vs_b  = alloc((size_t)M * HH * SVC);
    float* qp_b  = alloc((size_t)M * HH * PKC * 3);
    float* kp_b  = alloc((size_t)M * HH * PKC * 3);
    float* vp_b  = alloc((size_t)M * HH * PVC * 3);
    float* AQ    = alloc((size_t)BB * HH * NN * 32);
    float* AK    = alloc((size_t)BB * HH * NN * 32);
    float* vp_g  = alloc((size_t)M * HH * PVC * 3);
    float* pbias = alloc((size_t)BB * NN * NN * HH);
    float* attn  = alloc((size_t)BB * HH * NN * NN);
    float* rpt_g = alloc((size_t)BB * HH * NN * PVC * 3);
    float* obuf  = alloc((size_t)M * OIN);
    float* a_tmp = alloc((size_t)M * DIMC);
    float* h1    = alloc((size_t)M * DIMC);
    float* h2    = alloc((size_t)M * DIMC);
    float* updb  = alloc((size_t)M * 6);

    k_init<<<dim3((M * DIMC + 255) / 256), dim3(256), 0, stream>>>(xin, x, quats, trans, M * DIMC);

    for (int l = 0; l < DEPTH; ++l) {
        const float* Wsq_l   = Wsq   + (size_t)l * DIMC * 192;
        const float* Wsk_l   = Wsk   + (size_t)l * DIMC * 192;
        const float* Wsv_l   = Wsv   + (size_t)l * DIMC * 192;
        const float* Wpq_l   = Wpq   + (size_t)l * DIMC * 144;
        const float* Wpk_l   = Wpk   + (size_t)l * DIMC * 144;
        const float* Wpv_l   = Wpv   + (size_t)l * DIMC * 288;
        const float* ptw_l   = ptw   + (size_t)l * HH;
        const float* Wpair_l = Wpair + (size_t)l * PRC * HH;
        const float* bpair_l = bpair + (size_t)l * HH;
        const float* Wo_l    = Wo    + (size_t)l * OIN * DIMC;
        const float* bo_l    = bo    + (size_t)l * DIMC;
        const float* W1_l    = W1    + (size_t)l * DIMC * DIMC;
        const float* W2_l    = W2    + (size_t)l * DIMC * DIMC;
        const float* W3_l    = W3    + (size_t)l * DIMC * DIMC;
        const float* Wupd_l  = Wupd  + (size_t)l * DIMC * 6;

        k_compute_R<<<dim3(4), dim3(256), 0, stream>>>(quats, Rbuf);

        // projections: x (1024x512) @ W (512xNc)
        launch_gemm(stream, x, Wsq_l, qs_b, nullptr, M, 192, DIMC, DIMC, 192, 192,
                    0,0,1, 0,0,1, 0,0,1, 1, 0, 0);
        launch_gemm(stream, x, Wsk_l, ks_b, nullptr, M, 192, DIMC, DIMC, 192, 192,
                    0,0,1, 0,0,1, 0,0,1, 1, 0, 0);
        launch_gemm(stream, x, Wsv_l, vs_b, nullptr, M, 192, DIMC, DIMC, 192, 192,
                    0,0,1, 0,0,1, 0,0,1, 1, 0, 0);
        launch_gemm(stream, x, Wpq_l, qp_b, nullptr, M, 144, DIMC, DIMC, 144, 144,
                    0,0,1, 0,0,1, 0,0,1, 1, 0, 0);
        launch_gemm(stream, x, Wpk_l, kp_b, nullptr, M, 144, DIMC, DIMC, 144, 144,
                    0,0,1, 0,0,1, 0,0,1, 1, 0, 0);
        launch_gemm(stream, x, Wpv_l, vp_b, nullptr, M, 288, DIMC, DIMC, 288, 288,
                    0,0,1, 0,0,1, 0,0,1, 1, 0, 0);

        k_build_aqak<<<dim3(48), dim3(256), 0, stream>>>(qs_b, ks_b, qp_b, kp_b,
                                                         Rbuf, trans, ptw_l, AQ, AK);
        k_transform_vp<<<dim3(384), dim3(256), 0, stream>>>(vp_b, Rbuf, trans, vp_g);

        // pair bias: per (b,i): pair[b,i] (512x128) @ W_pair (128x12) + b_pair
        launch_gemm(stream, pair, Wpair_l, pbias, bpair_l, NN, HH, PRC, PRC, HH, HH,
                    0, (long)NN * PRC, M,   0, 0, 1,   0, (long)NN * HH, M,
                    M, 0, 0);

        // logits: per (b,h): AQ (512x32) @ AK^T -> (512x512)
        launch_gemm(stream, AQ, AK, attn, nullptr, NN, NN, 32, 32, 32, NN,
                    0, (long)NN * 32, BB * HH,
                    0, (long)NN * 32, BB * HH,
                    0, (long)NN * NN, BB * HH,
                    BB * HH, 1, 0);

        k_softmax<<<dim3(BB * HH * NN), dim3(256), 0, stream>>>(attn, pbias, mask);

        // rs = attn @ vs  -> o[:, h*16 .. ]
        launch_gemm(stream, attn, vs_b, obuf, nullptr, NN, SVC, NN, NN, HH * SVC, OIN,
                    0, (long)NN * NN, BB * HH,
                    (long)NN * HH * SVC, SVC, HH,
                    (long)NN * OIN, SVC, HH,
                    BB * HH, 0, 0);

        // rpt = attn @ vp_glob -> rpt_g (b,h,n,24)
        launch_gemm(stream, attn, vp_g, rpt_g, nullptr, NN, PVC * 3, NN, NN, HH * PVC * 3, PVC * 3,
                    0, (long)NN * NN, BB * HH,
                    (long)NN * HH * PVC * 3, PVC * 3, HH,
                    0, (long)NN * PVC * 3, BB * HH,
                    BB * HH, 0, 0);

        // rpair: per (b,i): attn[b,:,i,:] (12x512) @ pair[b,i] (512x128) -> o[:,576..]
        launch_gemm(stream, attn, pair, obuf + HH * SVC + HH * PVC * 4, nullptr,
                    HH, PRC, NN, NN * NN, PRC, PRC,
                    (long)HH * NN * NN, NN, NN,
                    0, (long)NN * PRC, M,
                    0, OIN, M,
                    M, 0, 0);

        k_finalize_rpt<<<dim3(384), dim3(256), 0, stream>>>(rpt_g, Rbuf, trans, obuf);

        // a = o @ W_o + b_o ; x = LN(a + x)
        launch_gemm(stream, obuf, Wo_l, a_tmp, bo_l, M, DIMC, OIN, OIN, DIMC, DIMC,
                    0,0,1, 0,0,1, 0,0,1, 1, 0, 0);
        k_layernorm<<<dim3(M), dim3(256), 0, stream>>>(a_tmp, x, ln1g + (size_t)l * DIMC,
                                                       ln1b + (size_t)l * DIMC);

        // FFN
        launch_gemm(stream, x,  W1_l, h1,    b1 + (size_t)l * DIMC, M, DIMC, DIMC,
                    DIMC, DIMC, DIMC, 0,0,1, 0,0,1, 0,0,1, 1, 0, 1);
        launch_gemm(stream, h1, W2_l, h2,    b2 + (size_t)l * DIMC, M, DIMC, DIMC,
                    DIMC, DIMC, DIMC, 0,0,1, 0,0,1, 0,0,1, 1, 0, 1);
        launch_gemm(stream, h2, W3_l, a_tmp, b3 + (size_t)l * DIMC, M, DIMC, DIMC,
                    DIMC, DIMC, DIMC, 0,0,1, 0,0,1, 0,0,1, 1, 0, 0);
        k_layernorm<<<dim3(M), dim3(256), 0, stream>>>(a_tmp, x, ln2g + (size_t)l * DIMC,
                                                       ln2b + (size_t)l * DIMC);

        // pose update head (512 -> 6)
        launch_gemm(stream, x, Wupd_l, updb, bupd + (size_t)l * 6, M, 6, DIMC,
                    DIMC, 6, 6, 0,0,1, 0,0,1, 0,0,1, 1, 0, 0);
        k_update_pose<<<dim3(4), dim3(256), 0, stream>>>(updb, Rbuf, quats, trans);
    }

    k_write_out<<<dim3((M * (DIMC + 7) + 255) / 256), dim3(256), 0, stream>>>(
        x, trans, quats, (float*)d_out);
}